// KernelMamba_43765716746827
// MI455X (gfx1250) — compile-verified
//
#include <hip/hip_runtime.h>
#include <math.h>

#define D_MODEL 1024
#define D_STATE 16
#define D_CONV  4
#define DT_RANK 64
#define D_INNER 2048
#define BATCH   2
#define SEQ     4096
#define NCHUNK  64
#define CHUNK_LEN (SEQ / NCHUNK)

typedef float v2f __attribute__((ext_vector_type(2)));
typedef float v8f __attribute__((ext_vector_type(8)));
typedef unsigned int u32x4 __attribute__((ext_vector_type(4)));
typedef int i32x4 __attribute__((ext_vector_type(4)));
typedef int i32x8 __attribute__((ext_vector_type(8)));

__device__ __forceinline__ float silu_f(float x) { return x / (1.0f + __expf(-x)); }

// ---------------------------------------------------------------------------
// Tensor Data Mover: async 2-D tile load global -> LDS.
// D# built per CDNA5 ISA. Tensor dims are passed as the
// *remaining* extent from the tile origin so edge tiles zero-fill (OOB reads
// return zero). data_size=4B. pad_interval/pad_amount inject LDS row padding.
// This toolchain exposes the 6-arg builtin:
//   (u32x4 g0, i32x8 g1, i32x4 g2, i32x4 g3, i32x8 extra, i32 cpol)
// ---------------------------------------------------------------------------
__device__ __forceinline__ void tdm_load_tile(
    unsigned lds_byte_off, const float* tile_ptr,
    unsigned rem_w, unsigned rem_h, unsigned long long row_stride_elems,
    unsigned tile_w, unsigned tile_h, unsigned pad_interval, unsigned pad_amount)
{
  unsigned long long ga = (unsigned long long)(uintptr_t)tile_ptr;
  u32x4 g0;
  g0[0] = 1u;                                              // count=1, user mode
  g0[1] = lds_byte_off;                                    // lds_addr (bytes)
  g0[2] = (unsigned)ga;                                    // global_addr[31:0]
  g0[3] = (unsigned)((ga >> 32) & 0x01ffffffu) | (2u << 30); // addr[56:32] | type=2
  i32x8 g1;
  g1[0] = (int)((2u << 16)                                 // data_size = 4 bytes
              | (1u << 20)                                 // pad_enable
              | (pad_interval << 22) | (pad_amount << 25));
  g1[1] = (int)((rem_w & 0xffffu) << 16);                  // tensor_dim0[15:0]
  g1[2] = (int)((rem_w >> 16) | ((rem_h & 0xffffu) << 16));// dim0[31:16]|dim1[15:0]
  g1[3] = (int)((rem_h >> 16) | (tile_w << 16));           // dim1[31:16]|tile_dim0
  g1[4] = (int)tile_h;                                     // tile_dim1 (tile_dim2=0)
  g1[5] = (int)(unsigned)(row_stride_elems & 0xffffffffu); // tensor_dim0_stride lo
  g1[6] = (int)(unsigned)((row_stride_elems >> 32) & 0xffffu); // stride hi (dim1_stride=0)
  g1[7] = 0;
  i32x4 z4 = {0, 0, 0, 0};
  i32x8 z8 = {0, 0, 0, 0, 0, 0, 0, 0};
  __builtin_amdgcn_tensor_load_to_lds(g0, g1, z4, z4, z8, 0);
}

// ---------------------------------------------------------------------------
// Generic f32 GEMM: C[M,N] = A[M,K](lda) * B[K,N](ldb), V_WMMA_F32_16X16X4_F32.
// TDM double-buffered staging: wave0 DMAs the next A/B tiles to LDS while all
// 8 waves (4M x 2N, 2x4 WMMA tiles each) compute on the current buffer.
// LDS: A rows padded to 17 dwords; B rows to 136 dwords (2*136 mod 64 = 16 ->
// lane groups 0-15 / 16-31 of a fragment read hit disjoint bank ranges).
// mode 0: plain store; mode 1: softplus(acc + bias[col]).
// ---------------------------------------------------------------------------
#define BM 128
#define BN 128
#define BK 16
#define A_STRIDE 17
#define B_STRIDE 136

__global__ __launch_bounds__(256) void gemm_f32_wmma(
    const float* __restrict__ A, int lda,
    const float* __restrict__ Bm, int ldb,
    float* __restrict__ C, int ldc,
    int M, int N, int K,
    const float* __restrict__ bias, int mode)
{
  __shared__ float As[2][BM * A_STRIDE];   // 2 x 8704 B
  __shared__ float Bs[2][BK * B_STRIDE];   // 2 x 8704 B

  const int tid   = threadIdx.x;
  const int lane  = tid & 31;
  const int wid   = tid >> 5;
  const int waveM = wid & 3;
  const int waveN = wid >> 2;
  const int rowBase = blockIdx.y * BM;
  const int colBase = blockIdx.x * BN;
  const int ln16 = lane & 15;
  const int khi  = (lane >> 4) << 1;       // lanes 16-31 hold K+2,K+3 (16x4 f32 A layout)

  v8f acc[2][4] = {};

  // Prologue: wave0 launches DMA for tile 0 into buffer 0.
  if (wid == 0) {
    tdm_load_tile((unsigned)(uintptr_t)&As[0][0], A + (size_t)rowBase * lda,
                  (unsigned)K, (unsigned)(M - rowBase), (unsigned long long)lda,
                  BK, BM, /*pad 16->17*/ 3, 0);
    tdm_load_tile((unsigned)(uintptr_t)&Bs[0][0], Bm + colBase,
                  (unsigned)(N - colBase), (unsigned)K, (unsigned long long)ldb,
                  BN, BK, /*pad 128->136*/ 6, 7);
  }

  int it = 0;
  for (int k0 = 0; k0 < K; k0 += BK, ++it) {
    const int cur = it & 1;
    if (wid == 0) {
      const bool more = (k0 + BK) < K;
      if (more) {
        const int kn = k0 + BK;
        tdm_load_tile((unsigned)(uintptr_t)&As[cur ^ 1][0],
                      A + (size_t)rowBase * lda + kn,
                      (unsigned)(K - kn), (unsigned)(M - rowBase),
                      (unsigned long long)lda, BK, BM, 3, 0);
        tdm_load_tile((unsigned)(uintptr_t)&Bs[cur ^ 1][0],
                      Bm + (size_t)kn * ldb + colBase,
                      (unsigned)(N - colBase), (unsigned)(K - kn),
                      (unsigned long long)ldb, BN, BK, 6, 7);
        __builtin_amdgcn_s_wait_tensorcnt(2);  // current pair done, next pair in flight
      } else {
        __builtin_amdgcn_s_wait_tensorcnt(0);  // drain
      }
    }
    __syncthreads();                            // publish LDS tile to all waves

    const float* Asb = &As[cur][0];
    const float* Bsb = &Bs[cur][0];
#pragma unroll
    for (int kk = 0; kk < BK; kk += 4) {
      v2f afrag[2], bfrag[4];
#pragma unroll
      for (int mt = 0; mt < 2; mt++) {
        int r = waveM * 32 + mt * 16 + ln16;
        afrag[mt].x = Asb[r * A_STRIDE + kk + khi];
        afrag[mt].y = Asb[r * A_STRIDE + kk + khi + 1];
      }
#pragma unroll
      for (int nt = 0; nt < 4; nt++) {
        int c = waveN * 64 + nt * 16 + ln16;
        bfrag[nt].x = Bsb[(kk + khi) * B_STRIDE + c];
        bfrag[nt].y = Bsb[(kk + khi + 1) * B_STRIDE + c];
      }
#pragma unroll
      for (int mt = 0; mt < 2; mt++)
#pragma unroll
        for (int nt = 0; nt < 4; nt++)
          acc[mt][nt] = __builtin_amdgcn_wmma_f32_16x16x4_f32(
              false, afrag[mt], false, bfrag[nt], (short)0, acc[mt][nt], false, false);
    }
    __syncthreads();  // all waves done reading buf[cur] before it is re-filled
  }

  // C/D layout: VGPR r -> row r (lanes 0-15) / row r+8 (lanes 16-31), col = lane%16
#pragma unroll
  for (int mt = 0; mt < 2; mt++) {
#pragma unroll
    for (int nt = 0; nt < 4; nt++) {
      int col = colBase + waveN * 64 + nt * 16 + ln16;
      if (col >= N) continue;
      int rbase = rowBase + waveM * 32 + mt * 16 + ((lane >> 4) << 3);
#pragma unroll
      for (int r = 0; r < 8; r++) {
        int row = rbase + r;
        if (row >= M) continue;
        float v = acc[mt][nt][r];
        if (mode == 1) {
          v += bias[col];
          v = (v > 20.0f) ? v : log1pf(__expf(v));
        }
        C[(size_t)row * ldc + col] = v;
      }
    }
  }
}

// ---------------------------------------------------------------------------
// Causal depthwise conv (D_CONV=4) + bias + SiLU.  u lives in xz[:, 0:D_INNER].
// ---------------------------------------------------------------------------
__global__ __launch_bounds__(256) void conv_silu_kernel(
    const float* __restrict__ xz, const float* __restrict__ conv_w,
    const float* __restrict__ conv_b, float* __restrict__ u_c)
{
  size_t g = (size_t)blockIdx.x * 256 + threadIdx.x;
  int d = (int)(g % D_INNER);
  size_t bl = g / D_INNER;            // b*L + l
  int l = (int)(bl % SEQ);
  float acc = conv_b[d];
  const float* w = conv_w + d * D_CONV;
#pragma unroll
  for (int j = 0; j < D_CONV; j++) {
    int t = l - (D_CONV - 1) + j;
    if (t >= 0)
      acc += xz[(bl + (size_t)(t - l)) * (size_t)(2 * D_INNER) + d] * w[j];
  }
  u_c[bl * D_INNER + d] = silu_f(acc);
}

// ---------------------------------------------------------------------------
// Chunked selective scan. Recurrence per (b,d,n): h = exp(dt*A)*h + dt*u*B_t[n].
// Pass 1: per (b,d,chunk) compute chunk transition prod P and zero-state result S.
// ---------------------------------------------------------------------------
__global__ __launch_bounds__(256) void scan_pass1(
    const float* __restrict__ u_c, const float* __restrict__ dtb,
    const float* __restrict__ dbc, const float* __restrict__ A_log,
    float* __restrict__ P, float* __restrict__ S)
{
  int dblk = blockIdx.x & 7;
  int bc   = blockIdx.x >> 3;
  int c    = bc & (NCHUNK - 1);
  int b    = bc >> 6;
  int d    = dblk * 256 + threadIdx.x;

  float ac[D_STATE], p[D_STATE], s[D_STATE];
#pragma unroll
  for (int n = 0; n < D_STATE; n++) {
    ac[n] = -__expf(A_log[d * D_STATE + n]);   // A = -exp(A_log)
    p[n] = 1.0f; s[n] = 0.0f;
  }
  int t0 = c * CHUNK_LEN;
  for (int t = t0; t < t0 + CHUNK_LEN; t++) {
    size_t bl = (size_t)b * SEQ + t;
    float u  = u_c[bl * D_INNER + d];
    float dt = dtb[bl * D_INNER + d];
    float dtu = dt * u;
    const float* Bt = dbc + bl * 96 + DT_RANK;   // uniform per block -> scalar loads
#pragma unroll
    for (int n = 0; n < D_STATE; n++) {
      float a = __expf(dt * ac[n]);
      p[n] *= a;
      s[n] = s[n] * a + dtu * Bt[n];
    }
  }
  size_t off = (((size_t)b * NCHUNK + c) * D_INNER + d) * D_STATE;
#pragma unroll
  for (int n = 0; n < D_STATE; n++) { P[off + n] = p[n]; S[off + n] = s[n]; }
}

// Pass 2: per (b,d,n), sequentially combine 64 chunks; leave exclusive-prefix
// initial state h_in in S (in place).
__global__ __launch_bounds__(256) void scan_pass2(
    const float* __restrict__ P, float* __restrict__ S)
{
  size_t g = (size_t)blockIdx.x * 256 + threadIdx.x;   // over B*D_INNER*D_STATE
  int n = (int)(g & 15);
  int d = (int)((g >> 4) & (D_INNER - 1));
  int b = (int)(g >> 15);
  float h = 0.0f;
  for (int c = 0; c < NCHUNK; c++) {
    size_t off = (((size_t)b * NCHUNK + c) * D_INNER + d) * D_STATE + n;
    float pv = P[off];
    float sv = S[off];
    S[off] = h;                // h_in for this chunk
    h = pv * h + sv;
  }
}

// Pass 3: replay chunks from h_in, emit y, fuse D-skip + SiLU gate, and write
// gated result into the u-half of xz (so out_proj reads it with lda=2*D_INNER).
__global__ __launch_bounds__(256) void scan_pass3(
    const float* __restrict__ u_c, const float* __restrict__ dtb,
    const float* __restrict__ dbc, const float* __restrict__ A_log,
    const float* __restrict__ S, const float* __restrict__ Dp,
    float* __restrict__ xz)
{
  int dblk = blockIdx.x & 7;
  int bc   = blockIdx.x >> 3;
  int c    = bc & (NCHUNK - 1);
  int b    = bc >> 6;
  int d    = dblk * 256 + threadIdx.x;

  float ac[D_STATE], h[D_STATE];
  size_t off = (((size_t)b * NCHUNK + c) * D_INNER + d) * D_STATE;
#pragma unroll
  for (int n = 0; n < D_STATE; n++) {
    ac[n] = -__expf(A_log[d * D_STATE + n]);
    h[n] = S[off + n];
  }
  float dpv = Dp[d];
  int t0 = c * CHUNK_LEN;
  for (int t = t0; t < t0 + CHUNK_LEN; t++) {
    size_t bl = (size_t)b * SEQ + t;
    float u  = u_c[bl * D_INNER + d];
    float dt = dtb[bl * D_INNER + d];
    float dtu = dt * u;
    const float* Bt = dbc + bl * 96 + DT_RANK;
    const float* Ct = dbc + bl * 96 + DT_RANK + D_STATE;
    float y = 0.0f;
#pragma unroll
    for (int n = 0; n < D_STATE; n++) {
      float a = __expf(dt * ac[n]);
      h[n] = h[n] * a + dtu * Bt[n];
      y += h[n] * Ct[n];
    }
    float z = xz[bl * (size_t)(2 * D_INNER) + D_INNER + d];
    xz[bl * (size_t)(2 * D_INNER) + d] = (y + u * dpv) * silu_f(z);
  }
}

// ---------------------------------------------------------------------------
extern "C" void kernel_launch(void* const* d_in, const int* in_sizes, int n_in,
                              void* d_out, int out_size, void* d_ws, size_t ws_size,
                              hipStream_t stream)
{
  const float* x        = (const float*)d_in[0];
  const float* in_proj  = (const float*)d_in[1];
  const float* conv_w   = (const float*)d_in[2];
  const float* conv_b   = (const float*)d_in[3];
  const float* x_proj   = (const float*)d_in[4];
  const float* dt_proj  = (const float*)d_in[5];
  const float* dt_bias  = (const float*)d_in[6];
  const float* A_log    = (const float*)d_in[7];
  const float* Dp       = (const float*)d_in[8];
  const float* out_proj = (const float*)d_in[9];
  float* out = (float*)d_out;

  const size_t M = (size_t)BATCH * SEQ;        // 8192 rows
  // Workspace (floats): xz 33.5M | u_c 16.8M | dbc 0.79M | dt 16.8M | P 4.2M | S 4.2M
  float* ws   = (float*)d_ws;
  float* xz   = ws;  ws += M * (2 * D_INNER);
  float* u_c  = ws;  ws += M * D_INNER;
  float* dbc  = ws;  ws += M * 96;
  float* dtb  = ws;  ws += M * D_INNER;
  float* Pbuf = ws;  ws += (size_t)BATCH * NCHUNK * D_INNER * D_STATE;
  float* Sbuf = ws;  ws += (size_t)BATCH * NCHUNK * D_INNER * D_STATE;

  dim3 blk(256);

  // 1) xz = x @ in_proj_w                       [8192,1024]x[1024,4096]
  gemm_f32_wmma<<<dim3((2 * D_INNER) / BN, M / BM), blk, 0, stream>>>(
      x, D_MODEL, in_proj, 2 * D_INNER, xz, 2 * D_INNER,
      (int)M, 2 * D_INNER, D_MODEL, nullptr, 0);

  // 2) causal depthwise conv + SiLU -> u_c
  conv_silu_kernel<<<(unsigned)((M * D_INNER) / 256), blk, 0, stream>>>(
      xz, conv_w, conv_b, u_c);

  // 3) dbc = u_c @ x_proj_w                     [8192,2048]x[2048,96]
  gemm_f32_wmma<<<dim3(1, M / BM), blk, 0, stream>>>(
      u_c, D_INNER, x_proj, 96, dbc, 96,
      (int)M, 96, D_INNER, nullptr, 0);

  // 4) dt = softplus(dbc[:, :64] @ dt_proj_w + bias)   [8192,64]x[64,2048]
  gemm_f32_wmma<<<dim3(D_INNER / BN, M / BM), blk, 0, stream>>>(
      dbc, 96, dt_proj, D_INNER, dtb, D_INNER,
      (int)M, D_INNER, DT_RANK, dt_bias, 1);

  // 5-7) chunked selective scan (+fused skip & gate, written into xz u-half)
  scan_pass1<<<BATCH * NCHUNK * (D_INNER / 256), blk, 0, stream>>>(
      u_c, dtb, dbc, A_log, Pbuf, Sbuf);
  scan_pass2<<<(BATCH * D_INNER * D_STATE) / 256, blk, 0, stream>>>(Pbuf, Sbuf);
  scan_pass3<<<BATCH * NCHUNK * (D_INNER / 256), blk, 0, stream>>>(
      u_c, dtb, dbc, A_log, Sbuf, Dp, xz);

  // 8) out = y_gated @ out_proj_w               [8192,2048]x[2048,1024]
  gemm_f32_wmma<<<dim3(D_MODEL / BN, M / BM), blk, 0, stream>>>(
      xz, 2 * D_INNER, out_proj, D_MODEL, out, D_MODEL,
      (int)M, D_MODEL, D_INNER, nullptr, 0);
}